// YOLOLoss_51616916963422
// MI455X (gfx1250) — compile-verified
//
#include <hip/hip_runtime.h>
#include <math.h>

#define EPSF 1e-9f

namespace {
constexpr int B = 32, A = 3, H = 128, W = 128, C = 80, N = 50;
constexpr int S = B * A * H * W;   // 1,572,864 cells
constexpr int TPB = 256;           // 8 waves (wave32)
constexpr int NWG = 512;           // grid-stride: 12 cells per thread exactly
}

typedef float v2f __attribute__((ext_vector_type(2)));
typedef float v8f __attribute__((ext_vector_type(8)));

#if defined(__HIP_DEVICE_COMPILE__) && __has_builtin(__builtin_amdgcn_wmma_f32_16x16x4_f32)
#define USE_WMMA_REDUCE 1
#endif

__device__ __forceinline__ float lane_butterfly_sum(float v) {
  // wave32 full butterfly
  for (int off = 16; off > 0; off >>= 1) v += __shfl_xor(v, off, 32);
  return v;
}

// ---------------------------------------------------------------------------
// Kernel 1: init workspace (mask_conf = 0, cellmap = -1)
// ---------------------------------------------------------------------------
__global__ void yolo_init(float* __restrict__ mask_conf, int* __restrict__ cellmap) {
  int i = (blockIdx.x * blockDim.x + threadIdx.x) * 4;
  if (i < S) {  // S divisible by 4, 16B-aligned offsets
    float4 z = make_float4(0.f, 0.f, 0.f, 0.f);
    *reinterpret_cast<float4*>(mask_conf + i) = z;
    int4 m1; m1.x = -1; m1.y = -1; m1.z = -1; m1.w = -1;
    *reinterpret_cast<int4*>(cellmap + i) = m1;
  }
}

// ---------------------------------------------------------------------------
// Kernel 2: build masks. One lane per batch, sequential over targets so
// duplicate-cell writes are deterministic last-write-wins.
// ---------------------------------------------------------------------------
__global__ void yolo_scatter(const float* __restrict__ out_wh,
                             const float* __restrict__ target,
                             float* __restrict__ mask_conf,
                             int* __restrict__ cellmap) {
  int b = threadIdx.x;
  if (b >= B) return;
  const float* tb = target + (size_t)b * N * 5;
  for (int n = 0; n < N; ++n) {
    const float* t = tb + n * 5;
    float tx = t[0], ty = t[1], tw = t[2], th = t[3];
    if (!(tx >= 0.f)) break;  // cumprod validity: once invalid, all later invalid
    // jnp.round == round-half-even == rintf (default RTNE)
    int gx = min(max((int)rintf(tx * (float)W - 0.5f), 0), W - 1);
    int gy = min(max((int)rintf(ty * (float)H - 0.5f), 0), H - 1);
    float ious[A];
    float best = -1.0f;
    int anchor = 0;
    for (int a = 0; a < A; ++a) {
      int cell = ((b * A + a) * H + gy) * W + gx;
      float pw0 = out_wh[(size_t)cell * 2 + 0];
      float pw1 = out_wh[(size_t)cell * 2 + 1];
      float inter = fminf(tw, pw0) * fminf(th, pw1);
      float uni = tw * th + pw0 * pw1 - inter;
      float iou = inter / (uni + EPSF);
      ious[a] = iou;
      if (iou > best) { best = iou; anchor = a; }  // first max wins (strict >)
    }
    for (int a = 0; a < A; ++a)
      mask_conf[((b * A + a) * H + gy) * W + gx] = ious[a];
    cellmap[((b * A + anchor) * H + gy) * W + gx] = b * N + n;
  }
}

// ---------------------------------------------------------------------------
// Kernel 3: dense streaming reduction. Reads conf/mask_conf/cellmap densely;
// gathers out_xy/out_wh/out_class/target only at obj cells.
// Workgroup reduction via V_WMMA_F32_16X16X4_F32 (A = ones, K-dim sum,
// accumulated through C) + wave32 butterfly.
// ---------------------------------------------------------------------------
__global__ __launch_bounds__(TPB) void yolo_main_reduce(
    const float* __restrict__ out_xy, const float* __restrict__ out_wh,
    const float* __restrict__ out_conf, const float* __restrict__ out_class,
    const float* __restrict__ target, const float* __restrict__ mask_conf,
    const int* __restrict__ cellmap, float* __restrict__ partial) {
  const int tid = threadIdx.x;
  const int gid = blockIdx.x * TPB + tid;
  const int stride = NWG * TPB;
  const float noobj_bce = -(float)C * log1pf(EPSF);  // per non-obj cell (all C classes)

  float s_xy = 0.f, s_wh = 0.f, s_obj = 0.f, s_noobj = 0.f, s_bce = 0.f;

  for (int s = gid; s < S; s += stride) {
    float conf = out_conf[s];
    float mconf = mask_conf[s];
    int cm = cellmap[s];
    float d = conf - mconf;
    float d2 = d * d;
    bool obj = false;
    if (cm >= 0) {
      const float* t = target + (size_t)cm * 5;
      float tw0 = t[2];
      if (tw0 > 0.f) {  // mask_obj = mask_wh[...,0] > 0
        obj = true;
        s_obj += d2;
        v2f oxy = *reinterpret_cast<const v2f*>(out_xy + (size_t)s * 2);
        v2f owh = *reinterpret_cast<const v2f*>(out_wh + (size_t)s * 2);
        float dx = oxy.x - t[0], dy = oxy.y - t[1];
        s_xy += dx * dx + dy * dy;
        float w0 = sqrtf(owh.x) - sqrtf(tw0);
        float w1 = sqrtf(owh.y) - sqrtf(t[3]);
        s_wh += w0 * w0 + w1 * w1;
        int cls = (int)t[4];
        const float* oc = out_class + (size_t)s * C;
        float bacc = 0.f;
        for (int c = 0; c < C; ++c) {
          float p = oc[c];
          float lg = (c == cls) ? logf(p + EPSF) : log1pf(EPSF - p);
          bacc -= fmaxf(lg, -100.f);
        }
        s_bce += bacc;
      }
    }
    if (!obj) {
      s_noobj += d2;
      s_bce += noobj_bce;  // p==0, y==0 constant term for all C classes
    }
  }

  __shared__ float red[5][TPB];
  red[0][tid] = s_xy;  red[1][tid] = s_wh;  red[2][tid] = s_obj;
  red[3][tid] = s_noobj;  red[4][tid] = s_bce;
  __syncthreads();

  if (tid < 32) {  // wave 0 only: uniform branch, EXEC all ones inside
#ifdef USE_WMMA_REDUCE
    const v2f ones = {1.0f, 1.0f};
    for (int j = 0; j < 5; ++j) {
      v8f c = {};
      for (int i = 0; i < TPB; i += 64) {
        v2f bv;
        bv.x = red[j][i + tid];
        bv.y = red[j][i + 32 + tid];
        // D[m][n] = sum_k B[k][n] + C[m][n] : pure K-reduction, f32 precision
        c = __builtin_amdgcn_wmma_f32_16x16x4_f32(false, ones, false, bv,
                                                  (short)0, c, false, false);
      }
      // all rows of D identical; lanes 16-31 duplicate columns of lanes 0-15
      float tot = lane_butterfly_sum(c[0]) * 0.5f;
      if (tid == 0) partial[blockIdx.x * 5 + j] = tot;
    }
#else
    if (tid == 0) {
      for (int j = 0; j < 5; ++j) {
        float tot = 0.f;
        for (int i = 0; i < TPB; ++i) tot += red[j][i];
        partial[blockIdx.x * 5 + j] = tot;
      }
    }
#endif
  }
}

// ---------------------------------------------------------------------------
// Kernel 4: reduce NWG workgroup partials (WMMA chain) and combine loss.
// ---------------------------------------------------------------------------
__global__ void yolo_finalize(const float* __restrict__ partial,
                              float* __restrict__ out) {
  const int lane = threadIdx.x;  // launched with exactly 32 threads
  float tot[5];
#ifdef USE_WMMA_REDUCE
  const v2f ones = {1.0f, 1.0f};
  for (int j = 0; j < 5; ++j) {
    v8f c = {};
    for (int i = 0; i < NWG; i += 64) {
      v2f bv;
      bv.x = partial[(size_t)(i + lane) * 5 + j];
      bv.y = partial[(size_t)(i + 32 + lane) * 5 + j];
      c = __builtin_amdgcn_wmma_f32_16x16x4_f32(false, ones, false, bv,
                                                (short)0, c, false, false);
    }
    tot[j] = lane_butterfly_sum(c[0]) * 0.5f;
  }
#else
  for (int j = 0; j < 5; ++j) {
    float t = 0.f;
    for (int i = 0; i < NWG; ++i) t += partial[(size_t)i * 5 + j];
    tot[j] = t;
  }
#endif
  if (lane == 0) {
    const float invS = 1.0f / (float)S;
    // POS_W=5 (xy, wh over 2S elems), OBJ_W=1, NOOBJ_W=0.5 (over S),
    // CLASS_W=1 (over S*C)
    float loss = 5.0f * (tot[0] + tot[1]) * (0.5f * invS)
               + tot[2] * invS
               + 0.5f * tot[3] * invS
               + (tot[4] * invS) * (1.0f / (float)C);
    out[0] = loss;
  }
}

// ---------------------------------------------------------------------------
extern "C" void kernel_launch(void* const* d_in, const int* in_sizes, int n_in,
                              void* d_out, int out_size, void* d_ws, size_t ws_size,
                              hipStream_t stream) {
  (void)in_sizes; (void)n_in; (void)out_size; (void)ws_size;
  const float* out_xy    = (const float*)d_in[0];
  const float* out_wh    = (const float*)d_in[1];
  const float* out_conf  = (const float*)d_in[2];
  const float* out_class = (const float*)d_in[3];
  const float* target    = (const float*)d_in[4];
  float* out = (float*)d_out;

  float* ws        = (float*)d_ws;
  float* mask_conf = ws;                    // S floats
  int*   cellmap   = (int*)(ws + S);        // S ints
  float* partial   = ws + 2 * (size_t)S;    // NWG*5 floats

  int init_blocks = (S / 4 + TPB - 1) / TPB;
  yolo_init<<<init_blocks, TPB, 0, stream>>>(mask_conf, cellmap);
  yolo_scatter<<<1, 32, 0, stream>>>(out_wh, target, mask_conf, cellmap);
  yolo_main_reduce<<<NWG, TPB, 0, stream>>>(out_xy, out_wh, out_conf, out_class,
                                            target, mask_conf, cellmap, partial);
  yolo_finalize<<<1, 32, 0, stream>>>(partial, out);
}